// wt_18837726560396
// MI455X (gfx1250) — compile-verified
//
#include <hip/hip_runtime.h>

// 2x2 Haar wavelet (8,3,1080,1920) f32 -> (8,12,540,960) f32.
// Memory-bound streaming kernel (~398 MB moved); butterfly done with
// V_WMMA_F32_16X16X4_F32 (exact f32, D = Haar(16x4) x Pixels(4x16)).
// Non-temporal loads/stores since footprint (400MB) > L2 (192MB), zero reuse.

typedef __attribute__((ext_vector_type(2))) float v2f;
typedef __attribute__((ext_vector_type(8))) float v8f;

#define HAAR_S 0.25f          // 0.5 / FACT, FACT = 2.0
#define IN_H   1080
#define IN_W   1920
#define OUT_H  540
#define OUT_W  960
#define BC     24             // B*C = 8*3
#define TILES_PER_ROW   (OUT_W / 32)        // 30
#define TILES_PER_PLANE (OUT_H * TILES_PER_ROW)
#define TOTAL_TILES     (BC * TILES_PER_PLANE)  // 388800, exact

__global__ __launch_bounds__(256) void haar_wmma_kernel(const float* __restrict__ x,
                                                        float* __restrict__ y) {
    const int lane = threadIdx.x & 31;
    const int wave = blockIdx.x * (blockDim.x >> 5) + (threadIdx.x >> 5);

    // tile -> (bc plane, output row h, 32-wide column tile tw)
    int bc  = wave / TILES_PER_PLANE;
    int rem = wave - bc * TILES_PER_PLANE;
    int h   = rem / TILES_PER_ROW;
    int tw  = rem - h * TILES_PER_ROW;

    // ---- A matrix: 16x4 Haar rows * s, rows m>=4 zero ---------------------
    // A layout (32-bit 16x4): lanes 0-15 = M, VGPR0:{K=0|K=2}, VGPR1:{K=1|K=3}
    const int m  = lane & 15;
    const int hi = lane >> 4;         // 0: K=0,1 half  1: K=2,3 half
    v2f A;
    if (m < 4) {
        if (!hi) {                    // K=0 (a), K=1 (b) coefficients
            A.x = HAAR_S;
            A.y = (m & 1) ? -HAAR_S : HAAR_S;
        } else {                      // K=2 (c), K=3 (d) coefficients
            A.x = (m & 2) ? -HAAR_S : HAAR_S;
            A.y = ((m ^ (m >> 1)) & 1) ? -HAAR_S : HAAR_S;
        }
    } else {
        A.x = 0.0f; A.y = 0.0f;
    }

    // ---- B matrix loads: one float2 per lane, directly in WMMA layout -----
    // B layout (4x16): col N in lane N (K=0,1) and lane N+16 (K=2,3).
    // lanes 0-15 read input row 2h (a,b pairs); lanes 16-31 read row 2h+1 (c,d).
    const int n = lane & 15;
    const float* rowp = x + (long)bc * (IN_H * (long)IN_W)
                          + (long)(2 * h + hi) * IN_W;

    const int w0 = tw * 32 + n;       // sub-tile 0 output column
    v2f b0 = __builtin_nontemporal_load((const v2f*)(rowp + 2 * w0));
    v2f b1 = __builtin_nontemporal_load((const v2f*)(rowp + 2 * (w0 + 16)));

    v8f cz = {};
    v8f d0 = __builtin_amdgcn_wmma_f32_16x16x4_f32(false, A, false, b0,
                                                   (short)0, cz, false, false);
    v8f d1 = __builtin_amdgcn_wmma_f32_16x16x4_f32(false, A, false, b1,
                                                   (short)0, cz, false, false);

    // ---- pack two 16-pixel sub-tiles into full-wave 128B NT stores --------
    // D rows 0..3 (ll,lh,hl,hh) live in lanes 0-15, VGPR r. Move sub-tile 1's
    // lanes 0-15 into lanes 16-31 with a shuffle so stores are full-wave.
    const long planeStride = (long)OUT_H * OUT_W;
    float* outp = y + ((long)bc * 4) * planeStride
                    + (long)h * OUT_W + tw * 32 + lane;
    #pragma unroll
    for (int r = 0; r < 4; ++r) {
        float lo = d0[r];
        float hv = __shfl(d1[r], lane & 15, 32);
        float v  = (lane < 16) ? lo : hv;
        __builtin_nontemporal_store(v, outp + (long)r * planeStride);
    }
}

extern "C" void kernel_launch(void* const* d_in, const int* in_sizes, int n_in,
                              void* d_out, int out_size, void* d_ws, size_t ws_size,
                              hipStream_t stream) {
    const float* x = (const float*)d_in[0];
    float* y = (float*)d_out;
    // 388800 waves, 8 waves per 256-thread block -> 48600 blocks, exact cover
    dim3 grid(TOTAL_TILES / 8), block(256);
    hipLaunchKernelGGL(haar_wmma_kernel, grid, block, 0, stream, x, y);
}